// AttentionDecoder_58050777973361
// MI455X (gfx1250) — compile-verified
//
#include <hip/hip_runtime.h>
#include <math.h>
#include <stdint.h>

typedef __attribute__((ext_vector_type(2))) float v2f;
typedef __attribute__((ext_vector_type(8))) float v8f;

#define DEVINL __device__ __forceinline__

// ---------------------------------------------------------------------------
// Core WMMA wrapper: D(16x16,f32) = A(16x4,f32) * B(4x16,f32) + C
// 8-arg form: (neg_a, A, neg_b, B, c_mod, C, reuse_a, reuse_b)
// ---------------------------------------------------------------------------
DEVINL v8f wmma4(float ax, float ay, float bx, float by, v8f c) {
  v2f a; a.x = ax; a.y = ay;
  v2f b; b.x = bx; b.y = by;
  return __builtin_amdgcn_wmma_f32_16x16x4_f32(false, a, false, b, (short)0, c,
                                               false, false);
}

// ---------------------------------------------------------------------------
// D[M x out_d] = X[M x in_d] * W^T (W is (out_d, in_d) row-major) + bias
// ---------------------------------------------------------------------------
DEVINL void proj_gemm(const float* __restrict__ X, int sX,
                      const float* __restrict__ W, const float* __restrict__ bias,
                      int in_d, int out_d, int mtiles,
                      float* __restrict__ D, int sD) {
  const int ln = threadIdx.x;
  const int j = ln & 15, hi = ln >> 4;
  for (int mt = 0; mt < mtiles; ++mt) {
    const float* arow = X + (mt * 16 + j) * sX;
    for (int n0 = 0; n0 < out_d; n0 += 16) {
      const float* brow = W + (size_t)(n0 + j) * in_d;
      const float bb = bias[n0 + j];
      v8f acc0 = {bb, bb, bb, bb, bb, bb, bb, bb};
      v8f acc1 = {0.f, 0.f, 0.f, 0.f, 0.f, 0.f, 0.f, 0.f};
      for (int k0 = 0; k0 < in_d; k0 += 8) {
        const int ka = k0 + 2 * hi;
        const int kb = k0 + 4 + 2 * hi;
        acc0 = wmma4(arow[ka], arow[ka + 1], brow[ka], brow[ka + 1], acc0);
        acc1 = wmma4(arow[kb], arow[kb + 1], brow[kb], brow[kb + 1], acc1);
      }
#pragma unroll
      for (int i = 0; i < 8; ++i)
        D[(mt * 16 + i + 8 * hi) * sD + n0 + j] = acc0[i] + acc1[i];
    }
  }
}

// ---------------------------------------------------------------------------
// S[M x N] = scale * Q[M x dim] * K[N x dim]^T   (both LDS, row-major)
// ---------------------------------------------------------------------------
DEVINL void score_gemm(const float* __restrict__ Q, int sQ,
                       const float* __restrict__ K, int sK,
                       int dim, int mtiles, int ntiles, float scale,
                       float* __restrict__ S, int sS) {
  const int ln = threadIdx.x;
  const int j = ln & 15, hi = ln >> 4;
  for (int mt = 0; mt < mtiles; ++mt) {
    const float* arow = Q + (mt * 16 + j) * sQ;
    for (int nt = 0; nt < ntiles; ++nt) {
      const float* brow = K + (nt * 16 + j) * sK;
      v8f acc0 = {0.f, 0.f, 0.f, 0.f, 0.f, 0.f, 0.f, 0.f};
      v8f acc1 = acc0;
      for (int k0 = 0; k0 < dim; k0 += 8) {
        const int ka = k0 + 2 * hi;
        const int kb = k0 + 4 + 2 * hi;
        acc0 = wmma4(arow[ka], arow[ka + 1], brow[ka], brow[ka + 1], acc0);
        acc1 = wmma4(arow[kb], arow[kb + 1], brow[kb], brow[kb + 1], acc1);
      }
#pragma unroll
      for (int i = 0; i < 8; ++i)
        S[(mt * 16 + i + 8 * hi) * sS + nt * 16 + j] = (acc0[i] + acc1[i]) * scale;
    }
  }
}

// ---------------------------------------------------------------------------
// Y[M x out_d] = A[M x kdim] * V[kdim x out_d] + R[M x out_d]  (all LDS)
// ---------------------------------------------------------------------------
DEVINL void attend_gemm(const float* __restrict__ A, int sA,
                        const float* __restrict__ V, int sV,
                        const float* __restrict__ R, int sR,
                        int kdim, int mtiles, int out_d,
                        float* __restrict__ Y, int sY) {
  const int ln = threadIdx.x;
  const int j = ln & 15, hi = ln >> 4;
  for (int mt = 0; mt < mtiles; ++mt) {
    const float* arow = A + (mt * 16 + j) * sA;
    for (int n0 = 0; n0 < out_d; n0 += 16) {
      v8f acc0;
#pragma unroll
      for (int i = 0; i < 8; ++i)
        acc0[i] = R[(mt * 16 + i + 8 * hi) * sR + n0 + j];
      v8f acc1 = {0.f, 0.f, 0.f, 0.f, 0.f, 0.f, 0.f, 0.f};
      for (int k0 = 0; k0 < kdim; k0 += 8) {
        const int ka = k0 + 2 * hi;
        const int kb = k0 + 4 + 2 * hi;
        acc0 = wmma4(arow[ka], arow[ka + 1],
                     V[ka * sV + n0 + j], V[(ka + 1) * sV + n0 + j], acc0);
        acc1 = wmma4(arow[kb], arow[kb + 1],
                     V[kb * sV + n0 + j], V[(kb + 1) * sV + n0 + j], acc1);
      }
#pragma unroll
      for (int i = 0; i < 8; ++i)
        Y[(mt * 16 + i + 8 * hi) * sY + n0 + j] = acc0[i] + acc1[i];
    }
  }
}

DEVINL void softmax_rows(float* S, int sS, int rows, int cols, int colsPad) {
  for (int r = threadIdx.x; r < rows; r += 32) {
    float* row = S + r * sS;
    float mx = -1e30f;
    for (int c = 0; c < cols; ++c) mx = fmaxf(mx, row[c]);
    float sum = 0.f;
    for (int c = 0; c < cols; ++c) {
      float e = __expf(row[c] - mx);
      row[c] = e;
      sum += e;
    }
    const float inv = 1.f / sum;
    for (int c = 0; c < cols; ++c) row[c] *= inv;
    for (int c = cols; c < colsPad; ++c) row[c] = 0.f;
  }
}

DEVINL void layernorm_rows(const float* Y, int sY, int rows, int D,
                           const float* __restrict__ g,
                           const float* __restrict__ beta,
                           float* Dst, int sD) {
  for (int r = threadIdx.x; r < rows; r += 32) {
    const float* row = Y + r * sY;
    float s = 0.f;
    for (int d = 0; d < D; ++d) s += row[d];
    const float mean = s / (float)D;
    float v = 0.f;
    for (int d = 0; d < D; ++d) {
      const float t = row[d] - mean;
      v += t * t;
    }
    const float rstd = rsqrtf(v / (float)D + 1e-5f);
    float* drow = Dst + r * sD;
    for (int d = 0; d < D; ++d)
      drow[d] = (row[d] - mean) * rstd * g[d] + beta[d];
  }
}

DEVINL void load_rows128(float* dst, int stride, const float* __restrict__ src,
                         int rows) {
  const float4* s4 = (const float4*)src;
  const int n4 = rows * 32;  // 128 floats = 32 float4 per row
  for (int i = threadIdx.x; i < n4; i += 32) {
    const int r = i >> 5, c4 = i & 31;
    const float4 v = s4[i];
    float* d = dst + r * stride + c4 * 4;
    d[0] = v.x; d[1] = v.y; d[2] = v.z; d[3] = v.w;
  }
}

DEVINL void zero_rows(float* buf, int stride, int r0, int r1, int cols) {
  const int total = (r1 - r0) * cols;
  for (int i = threadIdx.x; i < total; i += 32)
    buf[(r0 + i / cols) * stride + (i % cols)] = 0.f;
}

// ---------------------------------------------------------------------------
// TDM async prefetch: 2D tile (20 rows x 128 f32) global -> LDS with LDS
// padding of 2 DWORDs after every 128 DWORDs (=> LDS row stride 130 floats).
// D# layout per CDNA5 ISA ch.8 (group0/group1); groups 2/3 zero (2D tensor,
// tile_dim2 = 0). Tracked by TENSORcnt.
// clang-23 (amdgpu-toolchain) builtin arity: 6 args
//   (u32x4 g0, i32x8 g1, i32x4 g2, i32x4 g3, i32x8 gx, i32 cpol)
// ---------------------------------------------------------------------------
#if __has_builtin(__builtin_amdgcn_tensor_load_to_lds)
#define HAVE_TDM 1
typedef __attribute__((ext_vector_type(4))) unsigned int u32x4;
typedef __attribute__((ext_vector_type(8))) int i32x8;
typedef __attribute__((ext_vector_type(4))) int i32x4;

DEVINL void tdm_prefetch_ts(const float* __restrict__ gsrc, unsigned lds_byte) {
  const unsigned long long ga = (unsigned long long)(uintptr_t)gsrc;
  u32x4 g0;
  g0[0] = 1u;                                          // count=1, user D#
  g0[1] = lds_byte;                                    // lds_addr [63:32]
  g0[2] = (unsigned)(ga & 0xFFFFFFFFull);              // global_addr [95:64]
  g0[3] = (unsigned)((ga >> 32) & 0x01FFFFFFull)       // global_addr [120:96]
          | (2u << 30);                                // type=2 ("image")
  i32x8 g1;
  g1[0] = (int)((2u << 16)        // data_size = 4 bytes
                | (1u << 20)      // pad_enable
                | (6u << 22)      // pad_interval: 128 DWORDs
                | (1u << 25));    // pad_amount: 2 DWORDs
  g1[1] = (int)(128u << 16);      // tensor_dim0 = 128 (bits [79:48], low part)
  g1[2] = (int)(20u << 16);       // tensor_dim0 hi=0 | tensor_dim1 = 20
  g1[3] = (int)(128u << 16);      // tensor_dim1 hi=0 | tile_dim0 = 128
  g1[4] = 20;                     // tile_dim1 = 20 | tile_dim2 = 0
  g1[5] = 128;                    // tensor_dim0_stride = 128 (low 32)
  g1[6] = 0;                      // stride hi | tensor_dim1_stride lo
  g1[7] = 0;
  i32x4 gz = {0, 0, 0, 0};
  i32x8 gz8 = {0, 0, 0, 0, 0, 0, 0, 0};
  __builtin_amdgcn_tensor_load_to_lds(g0, g1, gz, gz, gz8, 0);
}
#endif

DEVINL void wait_tensorcnt0() {
#if __has_builtin(__builtin_amdgcn_s_wait_tensorcnt)
  __builtin_amdgcn_s_wait_tensorcnt(0);
#else
  asm volatile("s_wait_tensorcnt 0x0" ::: "memory");
#endif
}

// ---------------------------------------------------------------------------
struct KParams {
  const float *current, *environment, *ts_in;
  const float *st_qw, *st_qb, *st_kw, *st_kb, *st_vw, *st_vb, *st_rw, *st_rb, *st_g, *st_beta;
  const float *tb_qw, *tb_qb, *tb_kw, *tb_kb, *tb_vw, *tb_vb, *tb_rw, *tb_rb, *tb_g, *tb_beta;
  const float *o1_qw, *o1_qb, *o1_kw, *o1_kb, *o1_vw, *o1_vb, *o1_rw, *o1_rb, *o1_g, *o1_beta;
  const float *o2_qw, *o2_qb, *o2_kw, *o2_kb, *o2_vw, *o2_vb, *o2_rw, *o2_rb, *o2_g, *o2_beta;
  const float *lin_w, *lin_b;
};

// One wave32 per batch element; ~64 KB LDS per workgroup.
__global__ void __launch_bounds__(32)
attn_decoder_kernel(KParams P, float* __restrict__ out) {
  const int b = blockIdx.x;
  const int ln = threadIdx.x;

  __shared__ __align__(16) float sX[2112];       // blkA input (16x132) / attend out
  __shared__ __align__(16) float sTS[32 * 130];  // ts tile (TDM prefetch dest)
  __shared__ __align__(16) float sYb[48 * 66];   // out_seq (16 state + 20 ts, pad 48)
  __shared__ __align__(16) float sP0[32 * 66];   // Q then V
  __shared__ __align__(16) float sP1[32 * 66];   // K then R
  __shared__ __align__(16) float sS[48 * 50];    // scores / attn
  __shared__ float sQd[48], sKd[48], sVd[48], sRd[48], sFlat[48];

  // -------- kick off async TDM prefetch of the ts tile (used in Block B) ---
#if defined(HAVE_TDM)
  tdm_prefetch_ts(P.ts_in + (size_t)b * 20 * 128, (unsigned)(uintptr_t)sTS);
#endif
  zero_rows(sTS, 130, 20, 32, 128);  // pad rows (disjoint from TDM dest rows)

  // ================= Block A: state attention (16 x 128 -> 16 x 64) ========
  load_rows128(sX, 132, P.current + (size_t)b * 15 * 128, 15);
  load_rows128(sX + 15 * 132, 132, P.environment + (size_t)b * 128, 1);
  __syncthreads();
  proj_gemm(sX, 132, P.st_qw, P.st_qb, 128, 64, 1, sP0, 66);
  proj_gemm(sX, 132, P.st_kw, P.st_kb, 128, 64, 1, sP1, 66);
  __syncthreads();
  score_gemm(sP0, 66, sP1, 66, 64, 1, 1, 0.125f, sS, 50);
  __syncthreads();
  softmax_rows(sS, 50, 16, 16, 16);
  __syncthreads();
  proj_gemm(sX, 132, P.st_vw, P.st_vb, 128, 64, 1, sP0, 66);  // V
  proj_gemm(sX, 132, P.st_rw, P.st_rb, 128, 64, 1, sP1, 66);  // residual
  __syncthreads();
  attend_gemm(sS, 50, sP0, 66, sP1, 66, 16, 1, 64, sX, 66);
  __syncthreads();
  layernorm_rows(sX, 66, 16, 64, P.st_g, P.st_beta, sYb, 66);
  __syncthreads();

  // ================= Block B: ts attention (20 x 128 -> 20 x 64) ===========
#if defined(HAVE_TDM)
  wait_tensorcnt0();                 // ts tile now resident in sTS
#else
  load_rows128(sTS, 130, P.ts_in + (size_t)b * 20 * 128, 20);
#endif
  __syncthreads();
  proj_gemm(sTS, 130, P.tb_qw, P.tb_qb, 128, 64, 2, sP0, 66);
  proj_gemm(sTS, 130, P.tb_kw, P.tb_kb, 128, 64, 2, sP1, 66);
  __syncthreads();
  score_gemm(sP0, 66, sP1, 66, 64, 2, 2, 0.125f, sS, 50);
  __syncthreads();
  softmax_rows(sS, 50, 32, 20, 32);  // mask: only 20 valid keys
  __syncthreads();
  proj_gemm(sTS, 130, P.tb_vw, P.tb_vb, 128, 64, 2, sP0, 66);
  proj_gemm(sTS, 130, P.tb_rw, P.tb_rb, 128, 64, 2, sP1, 66);
  __syncthreads();
  attend_gemm(sS, 50, sP0, 66, sP1, 66, 32, 2, 64, sX, 66);
  __syncthreads();
  layernorm_rows(sX, 66, 20, 64, P.tb_g, P.tb_beta, sYb + 16 * 66, 66);
  __syncthreads();

  // ================= Block C: out1 attention (36 x 64 -> 36 x 16) ==========
  zero_rows(sYb, 66, 36, 48, 64);  // pad out_seq rows to 48
  __syncthreads();
  proj_gemm(sYb, 66, P.o1_qw, P.o1_qb, 64, 16, 3, sP0, 18);
  proj_gemm(sYb, 66, P.o1_kw, P.o1_kb, 64, 16, 3, sP1, 18);
  __syncthreads();
  score_gemm(sP0, 18, sP1, 18, 16, 3, 3, 0.25f, sS, 50);
  __syncthreads();
  softmax_rows(sS, 50, 48, 36, 48);
  __syncthreads();
  proj_gemm(sYb, 66, P.o1_vw, P.o1_vb, 64, 16, 3, sP0, 18);
  proj_gemm(sYb, 66, P.o1_rw, P.o1_rb, 64, 16, 3, sP1, 18);
  __syncthreads();
  attend_gemm(sS, 50, sP0, 18, sP1, 18, 48, 3, 16, sX, 18);
  __syncthreads();
  layernorm_rows(sX, 18, 36, 16, P.o1_g, P.o1_beta, sX, 18);  // in place
  __syncthreads();

  // ================= Block D: out2 attention (out_d = 1) + final linear ====
  for (int r = ln; r < 36; r += 32) {
    const float* h = sX + r * 18;
    float q = P.o2_qb[0], k = P.o2_kb[0], v = P.o2_vb[0], res = P.o2_rb[0];
    for (int d = 0; d < 16; ++d) {
      const float x = h[d];
      q += x * P.o2_qw[d];
      k += x * P.o2_kw[d];
      v += x * P.o2_vw[d];
      res += x * P.o2_rw[d];
    }
    sQd[r] = q; sKd[r] = k; sVd[r] = v; sRd[r] = res;
  }
  __syncthreads();
  for (int r = ln; r < 36; r += 32) {
    const float q = sQd[r];
    float mx = -1e30f;
    for (int t = 0; t < 36; ++t) mx = fmaxf(mx, q * sKd[t]);
    float sum = 0.f, att = 0.f;
    for (int t = 0; t < 36; ++t) {
      const float e = __expf(q * sKd[t] - mx);
      sum += e;
      att += e * sVd[t];
    }
    const float y = att / sum + sRd[r];
    // LayerNorm over a size-1 dim, exactly as the reference computes it:
    const float mean = y;
    const float t0 = y - mean;          // == 0, but not foldable w/o fast-math
    const float var = t0 * t0;          // == 0
    sFlat[r] = t0 * rsqrtf(var + 1e-5f) * P.o2_g[0] + P.o2_beta[0];
  }
  __syncthreads();
  if (ln == 0) {
    float o0 = P.lin_b[0], o1 = P.lin_b[1];
    for (int s = 0; s < 36; ++s) {
      o0 += sFlat[s] * P.lin_w[s];
      o1 += sFlat[s] * P.lin_w[36 + s];
    }
    out[b] = o0;                                            // mu
    const float sp = (o1 > 20.f) ? o1 : log1pf(__expf(o1)); // softplus
    out[gridDim.x + b] = sp + 1e-6f;                        // sigma
  }
}

// ---------------------------------------------------------------------------
// Input order: JAX pytree flatten (dict keys sorted) of
// {current, environment, params{lin_b, lin_w, out1{...}, out2{...},
//  state{...}, ts{...}}, ts}; block leaves: beta,g,kb,kw,qb,qw,rb,rw,vb,vw.
// ---------------------------------------------------------------------------
extern "C" void kernel_launch(void* const* d_in, const int* in_sizes, int n_in,
                              void* d_out, int out_size, void* d_ws,
                              size_t ws_size, hipStream_t stream) {
  (void)in_sizes; (void)n_in; (void)out_size; (void)d_ws; (void)ws_size;
  auto F = [&](int i) { return (const float*)d_in[i]; };
  KParams P;
  P.current = F(0);
  P.environment = F(1);
  P.lin_b = F(2);
  P.lin_w = F(3);
  P.o1_beta = F(4);  P.o1_g = F(5);  P.o1_kb = F(6);  P.o1_kw = F(7);
  P.o1_qb = F(8);    P.o1_qw = F(9); P.o1_rb = F(10); P.o1_rw = F(11);
  P.o1_vb = F(12);   P.o1_vw = F(13);
  P.o2_beta = F(14); P.o2_g = F(15); P.o2_kb = F(16); P.o2_kw = F(17);
  P.o2_qb = F(18);   P.o2_qw = F(19); P.o2_rb = F(20); P.o2_rw = F(21);
  P.o2_vb = F(22);   P.o2_vw = F(23);
  P.st_beta = F(24); P.st_g = F(25); P.st_kb = F(26); P.st_kw = F(27);
  P.st_qb = F(28);   P.st_qw = F(29); P.st_rb = F(30); P.st_rw = F(31);
  P.st_vb = F(32);   P.st_vw = F(33);
  P.tb_beta = F(34); P.tb_g = F(35); P.tb_kb = F(36); P.tb_kw = F(37);
  P.tb_qb = F(38);   P.tb_qw = F(39); P.tb_rb = F(40); P.tb_rw = F(41);
  P.tb_vb = F(42);   P.tb_vw = F(43);
  P.ts_in = F(44);

  attn_decoder_kernel<<<dim3(16384), dim3(32), 0, stream>>>(P, (float*)d_out);
}